// GCN_3l_agg_37787122270452
// MI455X (gfx1250) — compile-verified
//
#include <hip/hip_runtime.h>
#include <math.h>

typedef __attribute__((ext_vector_type(2))) float v2f;
typedef __attribute__((ext_vector_type(8))) float v8f;

#define D_H 128
#define BN_EPS 1e-5f

__device__ __forceinline__ void atomic_add_f32(float* p, float v) {
    __hip_atomic_fetch_add(p, v, __ATOMIC_RELAXED, __HIP_MEMORY_SCOPE_AGENT);
}

// ---------------------------------------------------------------------------
// Degree accumulation: deg[col[e]] += 1  (self loop +1 folded in later)
// ---------------------------------------------------------------------------
__global__ void deg_kernel(const int* __restrict__ col, float* __restrict__ deg, int nE) {
    int i = blockIdx.x * blockDim.x + threadIdx.x;
    if (i < nE) atomic_add_f32(&deg[col[i]], 1.0f);
}

__global__ void dinv_kernel(float* __restrict__ deg, int nN) {
    int i = blockIdx.x * blockDim.x + threadIdx.x;
    if (i < nN) deg[i] = rsqrtf(deg[i] + 1.0f);   // +1 = self loop; deg>=1 always
}

// ---------------------------------------------------------------------------
// GEMM: out[nrows x 128] = A[nrows x 128] @ W[128 x 128], fp32 WMMA 16x16x4.
// One wave computes one 16x16 tile; 8 waves/block tile the 128 columns.
// nrows must be a multiple of 16 (N=50000 = 3125*16). EXEC all-ones.
// A layout (16x4 f32): lanes 0-15 M=0..15 K={k,k+1}; lanes 16-31 K={k+2,k+3}.
// B layout (4x16 f32): mirrored: VGPR0 = row k (lanes<16) / row k+2 (lanes>=16).
// D layout: VGPR r -> (M=r, N=lane) lanes<16, (M=r+8, N=lane-16) lanes>=16.
// ---------------------------------------------------------------------------
__global__ __launch_bounds__(256) void gemm128_kernel(const float* __restrict__ A,
                                                      const float* __restrict__ W,
                                                      float* __restrict__ out,
                                                      int nrows) {
    const int wave = threadIdx.x >> 5;        // column tile 0..7
    const int lane = threadIdx.x & 31;
    const int m    = lane & 15;
    const int hi   = lane >> 4;               // 0: K pair {k,k+1}; 1: {k+2,k+3}
    const int row  = blockIdx.x * 16 + m;
    const int n0   = wave * 16;

    const float* __restrict__ arow = A + (size_t)row * D_H;

    v8f acc = {};
#pragma unroll 4
    for (int k = 0; k < D_H; k += 4) {
        v2f a, b;
        a.x = arow[k + 2 * hi];
        a.y = arow[k + 2 * hi + 1];
        b.x = W[(size_t)(k + 2 * hi) * D_H + n0 + m];
        b.y = W[(size_t)(k + 2 * hi + 1) * D_H + n0 + m];
        acc = __builtin_amdgcn_wmma_f32_16x16x4_f32(false, a, false, b,
                                                    (short)0, acc, false, false);
    }

    const int colBase = n0 + m;
    const int rowBase = blockIdx.x * 16 + hi * 8;
#pragma unroll
    for (int r = 0; r < 8; ++r)
        out[(size_t)(rowBase + r) * D_H + colBase] = acc[r];
}

// ---------------------------------------------------------------------------
// Edge aggregation: agg[col] += h[row] * dinv[row]*dinv[col]
// One wave per edge; lane handles 4 contiguous features (float4 gather).
// Edges e in [nE, nE+nN) are the self loops (row=col=e-nE).
// ---------------------------------------------------------------------------
__global__ __launch_bounds__(256) void agg_kernel(const float* __restrict__ h,
                                                  const int* __restrict__ row,
                                                  const int* __restrict__ col,
                                                  const float* __restrict__ dinv,
                                                  float* __restrict__ agg,
                                                  int nE, int nN) {
    const int gid   = blockIdx.x * blockDim.x + threadIdx.x;
    const int lane  = gid & 31;
    const int nwave = (gridDim.x * blockDim.x) >> 5;
    const int total = nE + nN;

    for (int e = gid >> 5; e < total; e += nwave) {
        int r, c;
        if (e < nE) { r = row[e]; c = col[e]; }
        else        { r = e - nE; c = r; }
        const float nrm = dinv[r] * dinv[c];
        const float4 hv = *(const float4*)(h + (size_t)r * D_H + lane * 4);
        float* dst = agg + (size_t)c * D_H + lane * 4;
        atomic_add_f32(dst + 0, hv.x * nrm);
        atomic_add_f32(dst + 1, hv.y * nrm);
        atomic_add_f32(dst + 2, hv.z * nrm);
        atomic_add_f32(dst + 3, hv.w * nrm);
    }
}

// ---------------------------------------------------------------------------
// BN statistics: per-feature sum and sum-of-squares (coalesced, thread=feature)
// ---------------------------------------------------------------------------
__global__ __launch_bounds__(128) void bn_stats_kernel(const float* __restrict__ agg,
                                                       float* __restrict__ sums,
                                                       float* __restrict__ sumsq,
                                                       int nN) {
    const int f = threadIdx.x;
    float s = 0.0f, s2 = 0.0f;
    for (int r = blockIdx.x; r < nN; r += gridDim.x) {
        const float v = agg[(size_t)r * D_H + f];
        s += v;
        s2 += v * v;
    }
    atomic_add_f32(&sums[f], s);
    atomic_add_f32(&sumsq[f], s2);
}

// ---------------------------------------------------------------------------
// Fused BN (training-mode, biased var) + ReLU.  Per-layer bias b is a
// per-feature constant added before BN -> cancels against the mean; omitted.
// ---------------------------------------------------------------------------
__global__ __launch_bounds__(256) void bn_relu_kernel(const float* __restrict__ agg,
                                                      const float* __restrict__ gamma,
                                                      const float* __restrict__ beta,
                                                      const float* __restrict__ sums,
                                                      const float* __restrict__ sumsq,
                                                      float* __restrict__ out,
                                                      int nN) {
    const int idx = blockIdx.x * blockDim.x + threadIdx.x;
    if (idx >= nN * D_H) return;
    const int f = idx & (D_H - 1);
    const float invN = 1.0f / (float)nN;
    const float m    = sums[f] * invN;
    const float var  = sumsq[f] * invN - m * m;
    const float y    = gamma[f] * (agg[idx] - m) * rsqrtf(var + BN_EPS) + beta[f];
    out[idx] = fmaxf(y, 0.0f);
}

// ---------------------------------------------------------------------------
// Readout: out[n, 0..9] = [h1|h2|h3][n, :] @ Wf + bf.  One wave per node.
// ---------------------------------------------------------------------------
__global__ __launch_bounds__(256) void readout_kernel(const float* __restrict__ h1,
                                                      const float* __restrict__ h2,
                                                      const float* __restrict__ h3,
                                                      const float* __restrict__ Wf,
                                                      const float* __restrict__ bf,
                                                      float* __restrict__ out,
                                                      int nN) {
    const int gid  = blockIdx.x * blockDim.x + threadIdx.x;
    const int node = gid >> 5;
    const int lane = gid & 31;
    if (node >= nN) return;

    const float* __restrict__ r1 = h1 + (size_t)node * D_H;
    const float* __restrict__ r2 = h2 + (size_t)node * D_H;
    const float* __restrict__ r3 = h3 + (size_t)node * D_H;

    float acc[10];
#pragma unroll
    for (int c = 0; c < 10; ++c) acc[c] = 0.0f;

    for (int f = lane; f < D_H; f += 32) {
        const float a1 = r1[f], a2 = r2[f], a3 = r3[f];
        const float* __restrict__ w1 = Wf + (size_t)f * 10;
        const float* __restrict__ w2 = Wf + (size_t)(D_H + f) * 10;
        const float* __restrict__ w3 = Wf + (size_t)(2 * D_H + f) * 10;
#pragma unroll
        for (int c = 0; c < 10; ++c)
            acc[c] += a1 * w1[c] + a2 * w2[c] + a3 * w3[c];
    }

#pragma unroll
    for (int c = 0; c < 10; ++c)
#pragma unroll
        for (int off = 16; off > 0; off >>= 1)
            acc[c] += __shfl_xor(acc[c], off, 32);

    if (lane == 0) {
#pragma unroll
        for (int c = 0; c < 10; ++c)
            out[(size_t)node * 10 + c] = acc[c] + bf[c];
    }
}

// ---------------------------------------------------------------------------
static inline size_t align_up(size_t v, size_t a) { return (v + a - 1) & ~(a - 1); }

extern "C" void kernel_launch(void* const* d_in, const int* in_sizes, int n_in,
                              void* d_out, int out_size, void* d_ws, size_t ws_size,
                              hipStream_t stream) {
    // input order: x, edge_index, W1,b1, W2,b2, W3,b3, g1,be1, g2,be2, g3,be3, Wf,bf
    const float* x   = (const float*)d_in[0];
    const int*   ei  = (const int*)d_in[1];
    const float* W[3]  = { (const float*)d_in[2], (const float*)d_in[4], (const float*)d_in[6] };
    const float* g[3]  = { (const float*)d_in[8], (const float*)d_in[10], (const float*)d_in[12] };
    const float* be[3] = { (const float*)d_in[9], (const float*)d_in[11], (const float*)d_in[13] };
    const float* Wf  = (const float*)d_in[14];
    const float* bf  = (const float*)d_in[15];

    const int N = in_sizes[0] / D_H;
    const int E = in_sizes[1] / 2;
    const int* row = ei;       // sources
    const int* col = ei + E;   // destinations

    // workspace carve-out
    char* ws = (char*)d_ws;
    const size_t featBytes = (size_t)N * D_H * sizeof(float);
    float* dinv  = (float*)ws;  ws += align_up((size_t)N * sizeof(float), 256);
    float* sums  = (float*)ws;  ws += align_up(D_H * sizeof(float), 256);
    float* sumsq = (float*)ws;  ws += align_up(D_H * sizeof(float), 256);
    float* hlin  = (float*)ws;  ws += align_up(featBytes, 256);
    float* agg   = (float*)ws;  ws += align_up(featBytes, 256);
    float* h[3];
    for (int l = 0; l < 3; ++l) { h[l] = (float*)ws; ws += align_up(featBytes, 256); }

    // 1) symmetric GCN normalization
    hipMemsetAsync(dinv, 0, (size_t)N * sizeof(float), stream);
    deg_kernel<<<(E + 255) / 256, 256, 0, stream>>>(col, dinv, E);
    dinv_kernel<<<(N + 255) / 256, 256, 0, stream>>>(dinv, N);

    // 2) three GCN layers
    const float* hin = x;
    const int totalEdges = E + N;
    for (int l = 0; l < 3; ++l) {
        gemm128_kernel<<<N / 16, 256, 0, stream>>>(hin, W[l], hlin, N);
        hipMemsetAsync(agg, 0, featBytes, stream);
        agg_kernel<<<(totalEdges + 7) / 8, 256, 0, stream>>>(hlin, row, col, dinv, agg, E, N);
        hipMemsetAsync(sums, 0, D_H * sizeof(float), stream);
        hipMemsetAsync(sumsq, 0, D_H * sizeof(float), stream);
        bn_stats_kernel<<<512, 128, 0, stream>>>(agg, sums, sumsq, N);
        bn_relu_kernel<<<(N * D_H + 255) / 256, 256, 0, stream>>>(agg, g[l], be[l], sums, sumsq, h[l], N);
        hin = h[l];
    }

    // 3) readout
    readout_kernel<<<(N + 7) / 8, 256, 0, stream>>>(h[0], h[1], h[2], Wf, bf, (float*)d_out, N);
}